// GraphSAGE_73383811219521
// MI455X (gfx1250) — compile-verified
//
#include <hip/hip_runtime.h>

// ---------------------------------------------------------------------------
// GraphSAGE (2x mean-agg layers + linear head) for MI455X / gfx1250.
//
// Roofline: edge scatter/gather dominates (~3.3 GB vs 23.3 TB/s ≈ 140 us);
// agg[100000x128] f32 (51 MB) is L2-resident (192 MB L2) so f32 atomics run
// at L2 rate. GEMMs are only 1.5e10 FLOP -> fp32 WMMA (V_WMMA_F32_16X16X4_F32)
// is already non-binding, so we keep full fp32 precision instead of bf16.
//
// Scatter-adds use inline-asm GLOBAL_ATOMIC_ADD_F32 (non-returning, STOREcnt
// only) to guarantee the native L2 atomic path instead of a CAS retry loop.
// ---------------------------------------------------------------------------

typedef float v2f __attribute__((ext_vector_type(2)));
typedef float v8f __attribute__((ext_vector_type(8)));

constexpr int N_NODES   = 100000;
constexpr int N_EDGES   = 1600000;
constexpr int D         = 128;   // D_IN == D_HID
constexpr int N_CLASSES = 64;

// Native fire-and-forget f32 atomic add (ISA §15.18.3 op 86, GV mode).
// Non-returning (TH[0]=0) -> tracked on STOREcnt; s_endpgm's implicit
// wait-idle guarantees completion before the next kernel in the stream.
__device__ __forceinline__ void atomic_add_f32(float* p, float v) {
  asm volatile("global_atomic_add_f32 %0, %1, off"
               :
               : "v"(p), "v"(v)
               : "memory");
}

// ---- zero a float4 region ----
__global__ void zero_f4(float4* __restrict__ p, int n4) {
  int i = blockIdx.x * blockDim.x + threadIdx.x;
  if (i < n4) p[i] = make_float4(0.f, 0.f, 0.f, 0.f);
}

// ---- in-degree via f32 atomics ----
__global__ void degree_count(const int* __restrict__ dst, float* __restrict__ deg) {
  int e = blockIdx.x * blockDim.x + threadIdx.x;
  if (e < N_EDGES) atomic_add_f32(&deg[dst[e]], 1.0f);
}

// deg[i] <- 1 / max(deg[i], 1)   (reciprocal so GEMM kernel multiplies)
__global__ void make_rdeg(float* __restrict__ deg) {
  int i = blockIdx.x * blockDim.x + threadIdx.x;
  if (i < N_NODES) deg[i] = 1.0f / fmaxf(deg[i], 1.0f);
}

// ---- edge scatter-add: one wave per edge, float4 slice per lane ----
// gather h[src] (coalesced 512B row) -> 4x global_atomic_add_f32 into agg[dst]
__global__ __launch_bounds__(256)
void scatter_add(const float* __restrict__ h, const int* __restrict__ src,
                 const int* __restrict__ dst, float* __restrict__ agg) {
  int g    = blockIdx.x * blockDim.x + threadIdx.x;
  int e    = g >> 5;
  if (e >= N_EDGES) return;
  int lane = g & 31;
  int s = src[e], d = dst[e];
  const float4 v = *(const float4*)(h + (size_t)s * D + lane * 4);
  float* dp = agg + (size_t)d * D + lane * 4;
  atomic_add_f32(dp + 0, v.x);
  atomic_add_f32(dp + 1, v.y);
  atomic_add_f32(dp + 2, v.z);
  atomic_add_f32(dp + 3, v.w);
}

// ---------------------------------------------------------------------------
// Fused SAGE layer GEMM: out = act( A@W1 [+ (AGG*rdeg)@W2] + bias )
// One wave computes one 16x16 f32 output tile with V_WMMA_F32_16X16X4_F32.
//
// Fragment layouts (CDNA5 ISA 7.12.2):
//   A 16x4  f32: lane l -> row (l&15), K = 2*(l>>4) + {0,1}   (v2f)
//   B 4x16  f32: lane l -> col (l&15), K = 2*(l>>4) + {0,1}   (v2f)
//   C 16x16 f32: lane l -> col (l&15), VGPR v -> row v + 8*(l>>4)  (v8f)
// Two accumulators (self / neighbor) give independent WMMA dep-chains.
// ---------------------------------------------------------------------------
template <int NCOLS, bool RELU, bool HAS_NEIGH>
__global__ __launch_bounds__(256)
void sage_gemm(const float* __restrict__ A, const float* __restrict__ AGG,
               const float* __restrict__ RDEG,
               const float* __restrict__ W1, const float* __restrict__ W2,
               const float* __restrict__ bias, float* __restrict__ out) {
  constexpr int CT = NCOLS / 16;                       // column tiles
  const int lane = threadIdx.x & 31;
  const int tile = blockIdx.x * (blockDim.x >> 5) + (threadIdx.x >> 5);
  const int row0 = (tile / CT) * 16;
  const int n0   = (tile % CT) * 16;
  if (row0 >= N_NODES) return;                         // wave-uniform guard

  const int   sl = lane & 15;                          // A row / B,C column
  const int   kp = (lane >> 4) << 1;                   // K sub-offset 0 or 2
  const float rd = HAS_NEIGH ? RDEG[row0 + sl] : 0.f;
  const float* arow = A + (size_t)(row0 + sl) * D;
  const float* grow = HAS_NEIGH ? (AGG + (size_t)(row0 + sl) * D) : nullptr;

  v8f accS = {};
  v8f accN = {};

#if defined(__has_builtin) && __has_builtin(__builtin_amdgcn_wmma_f32_16x16x4_f32)
  #pragma unroll
  for (int kk = 0; kk < D; kk += 4) {
    v2f a;  a.x = arow[kk + kp];  a.y = arow[kk + kp + 1];
    v2f b;  b.x = W1[(kk + kp) * NCOLS + n0 + sl];
            b.y = W1[(kk + kp + 1) * NCOLS + n0 + sl];
    accS = __builtin_amdgcn_wmma_f32_16x16x4_f32(
        false, a, false, b, (short)0, accS, false, false);
    if constexpr (HAS_NEIGH) {
      v2f am; am.x = grow[kk + kp] * rd; am.y = grow[kk + kp + 1] * rd;
      v2f bn; bn.x = W2[(kk + kp) * NCOLS + n0 + sl];
              bn.y = W2[(kk + kp + 1) * NCOLS + n0 + sl];
      accN = __builtin_amdgcn_wmma_f32_16x16x4_f32(
          false, am, false, bn, (short)0, accN, false, false);
    }
  }
#else
  // Scalar fallback (should not trigger on gfx1250): each lane computes its
  // own 8 C elements directly.
  #pragma unroll 1
  for (int v = 0; v < 8; ++v) {
    const int m = v + ((lane >> 4) * 8);
    const float* ar = A + (size_t)(row0 + m) * D;
    const float* gr = HAS_NEIGH ? (AGG + (size_t)(row0 + m) * D) : nullptr;
    const float rdm = HAS_NEIGH ? RDEG[row0 + m] : 0.f;
    float s = 0.f, n = 0.f;
    for (int k = 0; k < D; ++k) {
      s += ar[k] * W1[k * NCOLS + n0 + sl];
      if (HAS_NEIGH) n += gr[k] * rdm * W2[k * NCOLS + n0 + sl];
    }
    accS[v] = s; accN[v] = n;
  }
#endif

  const float bv = bias[n0 + sl];
  const int   mh = (lane >> 4) * 8;
  #pragma unroll
  for (int v = 0; v < 8; ++v) {
    float val = accS[v] + (HAS_NEIGH ? accN[v] : 0.f) + bv;
    if (RELU) val = fmaxf(val, 0.f);
    out[(size_t)(row0 + mh + v) * NCOLS + n0 + sl] = val;
  }
}

// ---------------------------------------------------------------------------
extern "C" void kernel_launch(void* const* d_in, const int* in_sizes, int n_in,
                              void* d_out, int out_size, void* d_ws, size_t ws_size,
                              hipStream_t stream) {
  (void)in_sizes; (void)n_in; (void)out_size; (void)ws_size;

  const float* features = (const float*)d_in[0];
  const int*   src      = (const int*)d_in[1];
  const int*   dst      = (const int*)d_in[2];
  const float* Ws1      = (const float*)d_in[3];
  const float* Wn1      = (const float*)d_in[4];
  const float* b1       = (const float*)d_in[5];
  const float* Ws2      = (const float*)d_in[6];
  const float* Wn2      = (const float*)d_in[7];
  const float* b2       = (const float*)d_in[8];
  const float* Wout     = (const float*)d_in[9];
  const float* bout     = (const float*)d_in[10];
  float*       out      = (float*)d_out;

  // Workspace layout (all f32): deg(padded) | agg | h1 | h2  -> ~154 MB
  float* ws  = (float*)d_ws;
  float* deg = ws;                                    // 102400 floats (padded)
  float* agg = ws + 102400;                           // N_NODES * D
  float* h1  = agg + (size_t)N_NODES * D;
  float* h2  = h1 + (size_t)N_NODES * D;

  const int aggN4 = N_NODES * D / 4;                  // 3,200,000
  const int zN4   = (102400 + N_NODES * D) / 4;

  constexpr int ROW_TILES   = N_NODES / 16;           // 6250 (exact)
  constexpr int LYR_BLOCKS  = ROW_TILES * (D / 16) / 8;         // 6250
  constexpr int OUT_BLOCKS  = ROW_TILES * (N_CLASSES / 16) / 8; // 3125
  constexpr int SCAT_BLOCKS = N_EDGES * 32 / 256;               // 200000

  // Degrees (re-derived every call: ws is not persistent).
  zero_f4<<<(zN4 + 255) / 256, 256, 0, stream>>>((float4*)deg, zN4);
  degree_count<<<(N_EDGES + 255) / 256, 256, 0, stream>>>(dst, deg);
  make_rdeg<<<(N_NODES + 255) / 256, 256, 0, stream>>>(deg);

  // Layer 1: agg = segment_sum(features[src], dst); h1 = relu(...)
  scatter_add<<<SCAT_BLOCKS, 256, 0, stream>>>(features, src, dst, agg);
  sage_gemm<D, true, true><<<LYR_BLOCKS, 256, 0, stream>>>(
      features, agg, deg, Ws1, Wn1, b1, h1);

  // Layer 2
  zero_f4<<<(aggN4 + 255) / 256, 256, 0, stream>>>((float4*)agg, aggN4);
  scatter_add<<<SCAT_BLOCKS, 256, 0, stream>>>(h1, src, dst, agg);
  sage_gemm<D, true, true><<<LYR_BLOCKS, 256, 0, stream>>>(
      h1, agg, deg, Ws2, Wn2, b2, h2);

  // Classifier head: logits = h2 @ W_out + b_out
  sage_gemm<N_CLASSES, false, false><<<OUT_BLOCKS, 256, 0, stream>>>(
      h2, nullptr, nullptr, Wout, nullptr, bout, out);
}